// ConstituentAttention_4337916970048
// MI455X (gfx1250) — compile-verified
//
#include <hip/hip_runtime.h>

typedef float v2f __attribute__((ext_vector_type(2)));
typedef float v8f __attribute__((ext_vector_type(8)));

#define S_LEN 1024
#define BATCH 8
#define EMB   512
#define PDIM  64
#define QKDIM 128   // 2P

// ---------------------------------------------------------------------------
// Kernel 1: qk[b,s,:] = context[s,b,:] @ W^T + bias   via V_WMMA_F32_16X16X4_F32
// One wave per 16x16 output tile. M = S (per batch), N = 128, K = 512.
// A layout (16x4 f32): lane = 16*khalf + m ; v0/v1 hold K = 2*khalf + {0,1}
// B layout (4x16 f32): lane = 16*khalf + n ; v0/v1 hold K = 2*khalf + {0,1}
// C/D layout: VGPR r, lanes 0-15 -> M=r, lanes 16-31 -> M=r+8 ; N = lane&15
// ---------------------------------------------------------------------------
__global__ __launch_bounds__(256) void proj_wmma_kernel(
    const float* __restrict__ context,   // [S, B, E]
    const float* __restrict__ wmat,      // [128, 512]
    const float* __restrict__ bias,      // [128]
    float* __restrict__ qk)              // [B, S, 128]
{
    const int lane = threadIdx.x & 31;
    const int wave = (blockIdx.x * blockDim.x + threadIdx.x) >> 5;
    const int nt = wave & 7;          // 8 N-tiles
    const int mt = (wave >> 3) & 63;  // 64 M-tiles
    const int b  = wave >> 9;         // 8 batches

    const int llo   = lane & 15;
    const int khalf = lane >> 4;
    const int m0 = mt * 16;
    const int n0 = nt * 16;

    const float* arow = context + (size_t)(m0 + llo) * (BATCH * EMB) + (size_t)b * EMB;
    const float* brow = wmat + (size_t)(n0 + llo) * EMB;

    const float bv = bias[n0 + llo];
    v8f acc;
    #pragma unroll
    for (int r = 0; r < 8; ++r) acc[r] = bv;

    #pragma unroll 4
    for (int k = 0; k < EMB; k += 4) {
        v2f a  = *(const v2f*)(arow + k + 2 * khalf);
        v2f bb = *(const v2f*)(brow + k + 2 * khalf);
        acc = __builtin_amdgcn_wmma_f32_16x16x4_f32(
            /*neg_a=*/false, a, /*neg_b=*/false, bb,
            /*c_mod=*/(short)0, acc, /*reuse_a=*/false, /*reuse_b=*/false);
    }

    float* orow = qk + (size_t)b * (S_LEN * QKDIM) + (size_t)m0 * QKDIM + n0 + llo;
    #pragma unroll
    for (int r = 0; r < 8; ++r) {
        const int m = r + 8 * khalf;
        orow[(size_t)m * QKDIM] = acc[r];
    }
}

// ---------------------------------------------------------------------------
// Kernel 2: per batch — band scores, 2-way softmax, w[s], g[s], prefix sums G.
// One 1024-thread block per batch (thread s handles row s).
// ---------------------------------------------------------------------------
__global__ __launch_bounds__(1024) void band_kernel(
    const float* __restrict__ qk,     // [B,S,128]  (q = [:,:,0:64], k = [:,:,64:128])
    const float* __restrict__ prior,  // [B,S,S]
    float* __restrict__ wband,        // [B,S]
    float* __restrict__ Gpre)         // [B,S]
{
    __shared__ float sAup[S_LEN];
    __shared__ float sAdn[S_LEN];
    __shared__ float ssc[S_LEN];

    const int b = blockIdx.x;
    const int s = threadIdx.x;

    const float* base = qk + (size_t)b * S_LEN * QKDIM;
    const float* qrow = base + (size_t)s * QKDIM;

    float du = 0.f, dd = 0.f;
    if (s < S_LEN - 1) {
        const float* krow = base + (size_t)(s + 1) * QKDIM + PDIM;
        #pragma unroll 8
        for (int e = 0; e < PDIM; ++e) du += qrow[e] * krow[e];
        du *= (1.0f / (float)EMB);
    }
    if (s > 0) {
        const float* krow = base + (size_t)(s - 1) * QKDIM + PDIM;
        #pragma unroll 8
        for (int e = 0; e < PDIM; ++e) dd += qrow[e] * krow[e];
        dd *= (1.0f / (float)EMB);
    }

    // softmax over the (at most) two band entries; all masked entries underflow to 0
    float Aup, Adn;
    if (s == 0)              { Aup = 1.f; Adn = 0.f; }
    else if (s == S_LEN - 1) { Aup = 0.f; Adn = 1.f; }
    else {
        const float m  = fmaxf(du, dd);
        const float eu = __expf(du - m);
        const float ed = __expf(dd - m);
        const float z  = eu + ed;
        Aup = eu / z; Adn = ed / z;
    }
    sAup[s] = Aup; sAdn[s] = Adn;
    __syncthreads();

    float wv = 0.f, g = 0.f;
    if (s < S_LEN - 1) {
        wv = sqrtf(sAup[s] * sAdn[s + 1] + 1e-9f);
        const float p  = prior[(size_t)b * S_LEN * S_LEN + (size_t)s * S_LEN + (s + 1)];
        const float na = p + (1.f - p) * wv;
        g = __logf(na + 1e-9f);
    }
    wband[b * S_LEN + s] = wv;

    // Hillis–Steele inclusive scan of g, then shift to exclusive
    ssc[s] = g;
    __syncthreads();
    for (int off = 1; off < S_LEN; off <<= 1) {
        const float t = (s >= off) ? ssc[s - off] : 0.f;
        __syncthreads();
        ssc[s] += t;
        __syncthreads();
    }
    Gpre[b * S_LEN + s] = ssc[s] - g;   // G[s] = sum_{r<s} g[r]
}

// ---------------------------------------------------------------------------
// Kernel 3: elementwise over [B,S,S] — the memory-bound pass (~96 MB traffic).
// One 256-thread block per (b, i) row; each thread does 4 consecutive j (float4).
//   na[i,j] = p + (1-p)*v,  v = w[min] on band else sqrt(1e-9)
//   c [i,j] = (i==j) ? na : exp(G[max]-G[min]) + 1e-9
// ---------------------------------------------------------------------------
__global__ __launch_bounds__(256) void out_kernel(
    const float* __restrict__ prior,
    const float* __restrict__ wband,
    const float* __restrict__ Gpre,
    float* __restrict__ out_c,
    float* __restrict__ out_na)
{
    __shared__ float sG[S_LEN];
    const int row = blockIdx.x;            // b*S + i
    const int b = row >> 10;
    const int i = row & (S_LEN - 1);
    const int t = threadIdx.x;

    const float* Gb = Gpre + b * S_LEN;
    #pragma unroll
    for (int c = 0; c < 4; ++c) sG[t * 4 + c] = Gb[t * 4 + c];
    __syncthreads();

    const float Gi = sG[i];
    const size_t off = (size_t)row * S_LEN;
    const float4 p4 = ((const float4*)(prior + off))[t];
    const float pv[4] = {p4.x, p4.y, p4.z, p4.w};

    float cv[4], nav[4];
    #pragma unroll
    for (int c = 0; c < 4; ++c) {
        const int j = t * 4 + c;
        const float p = pv[c];
        const int d = j - i;
        const float v = (d == 1 || d == -1)
                      ? wband[b * S_LEN + ((d == 1) ? i : j)]
                      : 3.16227766e-5f;              // sqrt(1e-9)
        const float na = p + (1.f - p) * v;
        nav[c] = na;
        if (j == i) {
            cv[c] = na;
        } else {
            const float Gmin = (j > i) ? Gi : sG[j];
            const float Gmax = (j > i) ? sG[j] : Gi;
            cv[c] = __expf(Gmax - Gmin) + 1e-9f;
        }
    }
    const float4 c4  = {cv[0],  cv[1],  cv[2],  cv[3]};
    const float4 na4 = {nav[0], nav[1], nav[2], nav[3]};
    ((float4*)(out_c  + off))[t] = c4;
    ((float4*)(out_na + off))[t] = na4;
}

extern "C" void kernel_launch(void* const* d_in, const int* in_sizes, int n_in,
                              void* d_out, int out_size, void* d_ws, size_t ws_size,
                              hipStream_t stream) {
    const float* context = (const float*)d_in[0];  // [S,B,E]
    const float* prior   = (const float*)d_in[1];  // [B,S,S]
    const float* pw      = (const float*)d_in[2];  // [128,512]
    const float* pb      = (const float*)d_in[3];  // [128]
    float* out = (float*)d_out;                    // c_attn then neighbor_attn

    float* qk    = (float*)d_ws;                           // B*S*128 = 1,048,576 floats
    float* wband = qk + (size_t)BATCH * S_LEN * QKDIM;     // B*S
    float* Gpre  = wband + (size_t)BATCH * S_LEN;          // B*S

    // 4096 tiles (8 b * 64 mt * 8 nt), 1 wave each -> 512 blocks of 8 waves
    proj_wmma_kernel<<<512, 256, 0, stream>>>(context, pw, pb, qk);
    band_kernel<<<BATCH, 1024, 0, stream>>>(qk, prior, wband, Gpre);
    out_kernel<<<BATCH * S_LEN, 256, 0, stream>>>(
        prior, wband, Gpre, out, out + (size_t)BATCH * S_LEN * S_LEN);
}